// Subtractor48Bit_23407571764113
// MI455X (gfx1250) — compile-verified
//
#include <hip/hip_runtime.h>
#include <stdint.h>

// ---------------------------------------------------------------------------
// 48-bit binary subtractor, A - B via two's complement, bandwidth-bound.
// CDNA5 path: TDM tensor_load_to_lds double-buffered tile DMA + NT stores.
// All input traffic via the Tensor Data Mover; outputs via non-temporal
// global_store_b128. No matrix ops exist in this problem (bit-exact integer
// work, unrepresentable in fp32 WMMA accumulators), so wmma is intentionally 0.
// ---------------------------------------------------------------------------

typedef float    v4f __attribute__((ext_vector_type(4)));
typedef unsigned v4u __attribute__((ext_vector_type(4)));
typedef int      v4i __attribute__((ext_vector_type(4)));
typedef int      v8i __attribute__((ext_vector_type(8)));

#define ROW_BITS   48
#define ROW_BYTES  (ROW_BITS * 4)      // 192 B per row
#define TILE_ROWS  64                  // rows per TDM tile == threads per block
#define MAX_BLOCKS 4096

#if defined(__HIP_DEVICE_COMPILE__) && __has_builtin(__builtin_amdgcn_tensor_load_to_lds)
#define HAS_TDM 1
#else
#define HAS_TDM 0
#endif

// ---- per-row bit pack (floats {0,1} -> 48-bit integer), LSB first ----------
__device__ __forceinline__ unsigned long long pack_row(const float* __restrict__ r) {
  const v4f* r4 = (const v4f*)r;                  // rows are 192 B, 16 B aligned
  unsigned long long v = 0ull;
#pragma unroll
  for (int k = 0; k < 12; ++k) {
    v4f x = r4[k];
    unsigned nib = (unsigned)(x.x > 0.5f)
                 | ((unsigned)(x.y > 0.5f) << 1)
                 | ((unsigned)(x.z > 0.5f) << 2)
                 | ((unsigned)(x.w > 0.5f) << 3);
    v |= (unsigned long long)nib << (4 * k);
  }
  return v;
}

// ---- unpack difference bits + borrow, streamed with non-temporal stores ----
__device__ __forceinline__ void emit_row(unsigned long long ai, unsigned long long bi,
                                         float* __restrict__ outRow,
                                         float* __restrict__ borrowPtr) {
  // A + ~B + 1 (48-bit ripple adder) == (ai - bi) mod 2^48; borrow == (ai < bi)
  unsigned long long d = ai - bi;
  v4f* o4 = (v4f*)outRow;
#pragma unroll
  for (int k = 0; k < 12; ++k) {
    v4f o;
    o.x = ((d >> (4 * k + 0)) & 1ull) ? 1.0f : 0.0f;
    o.y = ((d >> (4 * k + 1)) & 1ull) ? 1.0f : 0.0f;
    o.z = ((d >> (4 * k + 2)) & 1ull) ? 1.0f : 0.0f;
    o.w = ((d >> (4 * k + 3)) & 1ull) ? 1.0f : 0.0f;
    __builtin_nontemporal_store(o, o4 + k);       // write-once stream: TH=NT
  }
  __builtin_nontemporal_store((ai < bi) ? 1.0f : 0.0f, borrowPtr);
}

#if HAS_TDM
__device__ __forceinline__ unsigned rfl(unsigned x) {
  return (unsigned)__builtin_amdgcn_readfirstlane((int)x);
}

// Issue one TDM 2-D tile load: tile_dim = [48, TILE_ROWS] f32 elements,
// tensor_dim0_stride = 48 (densely packed rows), destination = LDS tile.
__device__ __forceinline__ void tdm_load_tile(const float* __restrict__ gbase,
                                              long long rowStart,
                                              unsigned remainingRows,
                                              unsigned ldsByteOff) {
  unsigned long long gaddr =
      (unsigned long long)(uintptr_t)gbase + (unsigned long long)rowStart * ROW_BYTES;

  // ----- D# group 0: control / lds_addr / global_addr / type=2 -----
  v4u g0;
  g0.x = rfl(1u);                                             // count=1, user desc
  g0.y = rfl(ldsByteOff);                                     // lds_addr [63:32]
  g0.z = rfl((unsigned)(gaddr & 0xffffffffull));              // global_addr lo
  g0.w = rfl(((unsigned)(gaddr >> 32) & 0x01ffffffu) | 0x80000000u); // hi | type=2

  // ----- D# group 1: sizes / tile dims / strides -----
  v8i g1;
  g1[0] = (int)rfl(2u << 16);                       // data_size=2 (4 B), mask=0
  g1[1] = (int)rfl((unsigned)ROW_BITS << 16);       // tensor_dim0 = 48 (lo16 in [63:48])
  g1[2] = (int)rfl((remainingRows & 0xffffu) << 16);// dim0 hi=0 | tensor_dim1 lo16
  g1[3] = (int)rfl(((remainingRows >> 16) & 0xffffu)
                   | ((unsigned)ROW_BITS << 16));   // tensor_dim1 hi16 | tile_dim0=48
  g1[4] = (int)rfl((unsigned)TILE_ROWS);            // tile_dim1 | tile_dim2=0
  g1[5] = (int)rfl((unsigned)ROW_BITS);             // tensor_dim0_stride lo32 = 48
  g1[6] = 0;                                        // stride0 hi | stride1 lo
  g1[7] = 0;                                        // stride1 hi

  v4i z4 = {0, 0, 0, 0};
#if __clang_major__ >= 23
  v8i z8 = {0, 0, 0, 0, 0, 0, 0, 0};
  __builtin_amdgcn_tensor_load_to_lds(g0, g1, z4, z4, z8, 0);
#else
  __builtin_amdgcn_tensor_load_to_lds(g0, g1, z4, z4, 0);
#endif
}
#endif // HAS_TDM

__global__ __launch_bounds__(TILE_ROWS)
void sub48_kernel(const float* __restrict__ A, const float* __restrict__ B,
                  float* __restrict__ out, long long N, int nTiles) {
#if HAS_TDM
  __shared__ alignas(16) float ldsA[2][TILE_ROWS * ROW_BITS];   // 24 KB
  __shared__ alignas(16) float ldsB[2][TILE_ROWS * ROW_BITS];   // 24 KB

  const int tid = threadIdx.x;
  // Wave-uniform leader predicate via readfirstlane -> SGPR compare -> scalar
  // s_cbranch. Guarantees non-leader waves BRANCH AROUND the tensor ops
  // (TDM instructions issue regardless of EXEC, so exec-masking is not a
  // safe guard for them).
  const bool leader = (rfl((unsigned)tid) < 32u);

  const int t0 = blockIdx.x;
  const int stride = gridDim.x;
  if (t0 >= nTiles) return;

  // Prologue: leader wave kicks off DMA of tile t0 into buffer 0.
  if (leader) {
    long long r0 = (long long)t0 * TILE_ROWS;
    tdm_load_tile(A, r0, (unsigned)(N - r0), (unsigned)(uintptr_t)&ldsA[0][0]);
    tdm_load_tile(B, r0, (unsigned)(N - r0), (unsigned)(uintptr_t)&ldsB[0][0]);
  }

  int buf = 0;
  for (int t = t0; t < nTiles; t += stride, buf ^= 1) {
    const int tn = t + stride;
    if (leader) {
      if (tn < nTiles) {
        // Prefetch next tile into the other buffer, then wait until only
        // those 2 prefetch DMAs remain outstanding -> current tile is ready.
        long long rn = (long long)tn * TILE_ROWS;
        tdm_load_tile(A, rn, (unsigned)(N - rn), (unsigned)(uintptr_t)&ldsA[buf ^ 1][0]);
        tdm_load_tile(B, rn, (unsigned)(N - rn), (unsigned)(uintptr_t)&ldsB[buf ^ 1][0]);
        __builtin_amdgcn_s_wait_tensorcnt(2);
      } else {
        __builtin_amdgcn_s_wait_tensorcnt(0);
      }
    }
    __syncthreads();                               // publish tile to all waves

    long long row = (long long)t * TILE_ROWS + tid;
    if (row < N) {
      unsigned long long ai = pack_row(&ldsA[buf][tid * ROW_BITS]);
      unsigned long long bi = pack_row(&ldsB[buf][tid * ROW_BITS]);
      emit_row(ai, bi, out + row * ROW_BITS, out + (long long)N * ROW_BITS + row);
    }
    __syncthreads();              // done reading buf before next DMA overwrites it
  }
#else
  // Fallback (also the path the host pass parses): direct global loads.
  for (int t = blockIdx.x; t < nTiles; t += gridDim.x) {
    long long row = (long long)t * TILE_ROWS + threadIdx.x;
    if (row < N) {
      unsigned long long ai = pack_row(A + row * ROW_BITS);
      unsigned long long bi = pack_row(B + row * ROW_BITS);
      emit_row(ai, bi, out + row * ROW_BITS, out + (long long)N * ROW_BITS + row);
    }
  }
#endif
}

extern "C" void kernel_launch(void* const* d_in, const int* in_sizes, int n_in,
                              void* d_out, int out_size, void* d_ws, size_t ws_size,
                              hipStream_t stream) {
  (void)n_in; (void)out_size; (void)d_ws; (void)ws_size;
  const float* A = (const float*)d_in[0];
  const float* B = (const float*)d_in[1];
  float* out = (float*)d_out;

  long long N = (long long)in_sizes[0] / ROW_BITS;
  if (N <= 0) return;
  int nTiles = (int)((N + TILE_ROWS - 1) / TILE_ROWS);
  int blocks = nTiles < MAX_BLOCKS ? nTiles : MAX_BLOCKS;

  sub48_kernel<<<dim3(blocks), dim3(TILE_ROWS), 0, stream>>>(A, B, out, N, nTiles);
}